// TimeWinEmbedding_8323646620555
// MI455X (gfx1250) — compile-verified
//
#include <hip/hip_runtime.h>
#include <hip/hip_bf16.h>

// Problem constants (match reference)
#define T_  8
#define B_  4096
#define V_  131072
#define E_  128
#define VV_ 50000
#define SV_ 100

#define SEG_PER_BLOCK 8      // segments (batch rows) per block; one wave32 per segment
#define MAIN_THREADS  256    // 8 waves
#define SCAN_THREADS  1024   // 4 counts per thread -> 4096 per window

#define TAB_FLOATS   (SV_ * E_)        // 12800 floats = 51200 B per window table
#define FILL_OPS     25                // 6400 float2 / 256 threads, exactly

// ---------------------------------------------------------------------------
// Kernel 1: exclusive prefix sum of ragged counts per window.
// offsets[t*B + b] = sum_{b'<b} counts[t*B + b']
// ---------------------------------------------------------------------------
__global__ __launch_bounds__(SCAN_THREADS)
void scan_offsets_kernel(const int* __restrict__ counts, int* __restrict__ offsets) {
    __shared__ int sdat[SCAN_THREADS];
    const int t    = blockIdx.x;
    const int tid  = threadIdx.x;
    const int base = t * B_;
    const int i0   = 4 * tid;

    int v0 = counts[base + i0 + 0];
    int v1 = counts[base + i0 + 1];
    int v2 = counts[base + i0 + 2];
    int v3 = counts[base + i0 + 3];
    int tot = v0 + v1 + v2 + v3;

    sdat[tid] = tot;
    __syncthreads();
    for (int d = 1; d < SCAN_THREADS; d <<= 1) {
        int x = sdat[tid];
        int y = (tid >= d) ? sdat[tid - d] : 0;
        __syncthreads();
        sdat[tid] = x + y;
        __syncthreads();
    }
    int excl = sdat[tid] - tot;

    offsets[base + i0 + 0] = excl;
    offsets[base + i0 + 1] = excl + v0;
    offsets[base + i0 + 2] = excl + v0 + v1;
    offsets[base + i0 + 3] = excl + v0 + v1 + v2;
}

// ---------------------------------------------------------------------------
// Async fill of one 51.2 KB source table into an LDS buffer.
// Every thread issues exactly FILL_OPS b64 async ops -> per-wave ASYNCcnt
// bookkeeping is uniform, enabling s_wait_asynccnt FILL_OPS for pipelining.
// ---------------------------------------------------------------------------
__device__ __forceinline__
void async_fill_table(float* __restrict__ dst, const float* __restrict__ stab, int tid) {
#if defined(__gfx1250__)
    const unsigned lds_base = (unsigned)(uintptr_t)dst;
    #pragma unroll
    for (int k = 0; k < FILL_OPS; ++k) {
        unsigned goff = (unsigned)((k * MAIN_THREADS + tid) * 8);  // float2 per op
        unsigned loff = lds_base + goff;
        asm volatile("global_load_async_to_lds_b64 %0, %1, %2"
                     :: "v"(loff), "v"(goff), "s"(stab)
                     : "memory");
    }
#else
    for (int k = tid; k < TAB_FLOATS / 2; k += MAIN_THREADS)
        ((float2*)dst)[k] = ((const float2*)stab)[k];
#endif
}

__device__ __forceinline__ void wait_async_le_fill() {
#if defined(__gfx1250__)
    asm volatile("s_wait_asynccnt 25" ::: "memory");   // older fill fully landed
#endif
}
__device__ __forceinline__ void wait_async_zero() {
#if defined(__gfx1250__)
    asm volatile("s_wait_asynccnt 0" ::: "memory");
#endif
}

// ---------------------------------------------------------------------------
// Kernel 2: fused gather*gather -> segment mean -> temporal weighted sum,
// with double-buffered async DMA of the per-window source table (LDS),
// value rows gathered from L2-resident tables with prefetch lookahead.
// One wave32 per segment; each lane owns 4 consecutive E elements.
// ---------------------------------------------------------------------------
__global__ __launch_bounds__(MAIN_THREADS)
void timewin_kernel(const int*   __restrict__ win_values,
                    const int*   __restrict__ win_sources,
                    const int*   __restrict__ counts,
                    const float* __restrict__ value_emb,
                    const float* __restrict__ source_emb,
                    const float* __restrict__ win_weight,
                    const int*   __restrict__ offsets,
                    float*       __restrict__ out) {
    __shared__ __align__(16) float s_src[2][TAB_FLOATS];   // 102,400 bytes

    const int tid    = threadIdx.x;
    const int lane_e = (tid & 31) * 4;                 // 4 consecutive E elems per lane
    const int j      = tid >> 5;                       // wave id -> segment within block
    const int b      = blockIdx.x * SEG_PER_BLOCK + j; // batch row

    float4 acc = make_float4(0.f, 0.f, 0.f, 0.f);

    // Prologue: start DMA of window 0's table.
    async_fill_table(&s_src[0][0], source_emb, tid);

    for (int t = 0; t < T_; ++t) {
        const int cur = t & 1;

        // Kick off next window's table DMA before waiting on the current one.
        if (t + 1 < T_) {
            async_fill_table(&s_src[cur ^ 1][0],
                             source_emb + (size_t)(t + 1) * TAB_FLOATS, tid);
            wait_async_le_fill();   // in-order ASYNCcnt: current buffer complete
        } else {
            wait_async_zero();
        }
        __syncthreads();            // current table visible to all waves

        const float* sbuf = &s_src[cur][0];
        const int cnt  = counts[t * B_ + b];
        const int off0 = offsets[t * B_ + b];
        const int*   vptr = win_values  + (size_t)t * V_ + off0;
        const int*   sptr = win_sources + (size_t)t * V_ + off0;
        const float* vtab = value_emb   + (size_t)t * ((size_t)VV_ * E_);

        float4 sum = make_float4(0.f, 0.f, 0.f, 0.f);
        int i = 0;
        for (; i + 1 < cnt; i += 2) {               // 2 value rows in flight
            int vi0 = vptr[i],     si0 = sptr[i];
            int vi1 = vptr[i + 1], si1 = sptr[i + 1];
            if (i + 3 < cnt) {                       // lookahead prefetch (b8)
                __builtin_prefetch(vtab + (size_t)vptr[i + 2] * E_ + lane_e, 0, 1);
                __builtin_prefetch(vtab + (size_t)vptr[i + 3] * E_ + lane_e, 0, 1);
            }
            const float4 vr0 = *(const float4*)(vtab + (size_t)vi0 * E_ + lane_e);
            const float4 vr1 = *(const float4*)(vtab + (size_t)vi1 * E_ + lane_e);
            const float4 sr0 = *(const float4*)(sbuf + si0 * E_ + lane_e);
            const float4 sr1 = *(const float4*)(sbuf + si1 * E_ + lane_e);
            sum.x = fmaf(vr0.x, sr0.x, sum.x);
            sum.y = fmaf(vr0.y, sr0.y, sum.y);
            sum.z = fmaf(vr0.z, sr0.z, sum.z);
            sum.w = fmaf(vr0.w, sr0.w, sum.w);
            sum.x = fmaf(vr1.x, sr1.x, sum.x);
            sum.y = fmaf(vr1.y, sr1.y, sum.y);
            sum.z = fmaf(vr1.z, sr1.z, sum.z);
            sum.w = fmaf(vr1.w, sr1.w, sum.w);
        }
        if (i < cnt) {                               // tail token
            int vi = vptr[i], si = sptr[i];
            const float4 vr = *(const float4*)(vtab + (size_t)vi * E_ + lane_e);
            const float4 sr = *(const float4*)(sbuf + si * E_ + lane_e);
            sum.x = fmaf(vr.x, sr.x, sum.x);
            sum.y = fmaf(vr.y, sr.y, sum.y);
            sum.z = fmaf(vr.z, sr.z, sum.z);
            sum.w = fmaf(vr.w, sr.w, sum.w);
        }

        const float w     = win_weight[t];
        const float fcnt  = (float)(cnt > 0 ? cnt : 1);
        const float scale = w / fcnt;                // scatter_mean + temporal weight
        acc.x = fmaf(sum.x, scale, acc.x);
        acc.y = fmaf(sum.y, scale, acc.y);
        acc.z = fmaf(sum.z, scale, acc.z);
        acc.w = fmaf(sum.w, scale, acc.w);

        __syncthreads();   // all reads of s_src[cur] done before it is refilled
    }

    *(float4*)(out + (size_t)b * E_ + lane_e) = acc;
}

// ---------------------------------------------------------------------------
// Host entry
// inputs: 0 win_values(T,V) i32 | 1 win_sources(T,V) i32 | 2 win_tokens_size(T,B) i32
//         3 value_emb(T,VV,E) f32 | 4 source_emb(T,SV,E) f32 | 5 win_weight(T) f32
// output: (B,E) f32
// ---------------------------------------------------------------------------
extern "C" void kernel_launch(void* const* d_in, const int* in_sizes, int n_in,
                              void* d_out, int out_size, void* d_ws, size_t ws_size,
                              hipStream_t stream) {
    const int*   win_values = (const int*)d_in[0];
    const int*   win_sources = (const int*)d_in[1];
    const int*   counts      = (const int*)d_in[2];
    const float* value_emb   = (const float*)d_in[3];
    const float* source_emb  = (const float*)d_in[4];
    const float* win_weight  = (const float*)d_in[5];
    float*       out         = (float*)d_out;
    int*         offsets     = (int*)d_ws;   // T_*B_ ints = 128 KB scratch

    scan_offsets_kernel<<<T_, SCAN_THREADS, 0, stream>>>(counts, offsets);
    timewin_kernel<<<B_ / SEG_PER_BLOCK, MAIN_THREADS, 0, stream>>>(
        win_values, win_sources, counts, value_emb, source_emb, win_weight,
        offsets, out);
}